// AittalaGCN1dBlock_11751030522225
// MI455X (gfx1250) — compile-verified
//
#include <hip/hip_runtime.h>
#include <hip/hip_bf16.h>

typedef __attribute__((ext_vector_type(2))) float v2f;
typedef __attribute__((ext_vector_type(4))) float v4f;
typedef __attribute__((ext_vector_type(8))) float v8f;
typedef __attribute__((ext_vector_type(4))) int v4i;

typedef __attribute__((address_space(1))) v4i* as1v4i_t;
typedef __attribute__((address_space(3))) v4i* as3v4i_t;

namespace {

constexpr int kB = 4, kN = 16, kG = 64, kL = 2048, kE = 32768;
constexpr int kET = kE + kL;      // edges incl. self loops = 34816
constexpr int kF = 128;           // GCN output channels
constexpr int kC2 = 256;          // after concat with group-max
constexpr int kM = kG * kL;       // 131072 GEMM rows
constexpr float kEps = 1e-5f;

// LDS tile strides (padded for bank-conflict-free 16x16x4 lane access)
constexpr int kAStride = 68;      // 16 rows x 64 k  (4-bank spread per row)
constexpr int kBStride = 136;     // 64 k x 128 cols (2*136 % 64 == 16)

// 16-byte global->LDS copy; async (ASYNCcnt-tracked) when the builtin exists.
__device__ inline void cp16(const float* gp, float* lp) {
#if __has_builtin(__builtin_amdgcn_global_load_async_to_lds_b128)
  __builtin_amdgcn_global_load_async_to_lds_b128((as1v4i_t)gp, (as3v4i_t)lp, 0, 0);
#else
  *(v4f*)lp = *(const v4f*)gp;
#endif
}

__device__ inline void cp_wait() {
#if __has_builtin(__builtin_amdgcn_global_load_async_to_lds_b128) && \
    __has_builtin(__builtin_amdgcn_s_wait_asynccnt)
  __builtin_amdgcn_s_wait_asynccnt(0);
#endif
}

// x[B,N,64,L] -> h[(g*L+l)*64 + c]
__global__ void k_transpose_x(const float* __restrict__ x, float* __restrict__ h) {
  int tid = blockIdx.x * blockDim.x + threadIdx.x;      // total kG*kL*64
  int c = tid & 63;
  int l = (tid >> 6) & (kL - 1);
  int g = tid >> 17;
  h[tid] = x[((size_t)g * 64 + c) * kL + l];
}

__global__ void k_deg_init(float* deg) {
  int l = blockIdx.x * blockDim.x + threadIdx.x;
  if (l < kL) deg[l] = 2.0f;      // improved=True self-loop weight
}

__global__ void k_deg_acc(const long long* __restrict__ ei, float* deg) {
  int e = blockIdx.x * blockDim.x + threadIdx.x;
  if (e < kE) atomicAdd(&deg[(int)ei[kE + e]], 1.0f);
}

__global__ void k_dinv(float* deg) {
  int l = blockIdx.x * blockDim.x + threadIdx.x;
  if (l < kL) { float d = deg[l]; deg[l] = d > 0.f ? rsqrtf(d) : 0.f; }
}

__global__ void k_norm(const long long* __restrict__ ei, const float* __restrict__ dinv,
                       float* __restrict__ nrm) {
  int e = blockIdx.x * blockDim.x + threadIdx.x;
  if (e >= kET) return;
  if (e < kE) {
    nrm[e] = dinv[(int)ei[e]] * dinv[(int)ei[kE + e]];
  } else {
    float d = dinv[e - kE];
    nrm[e] = 2.0f * d * d;
  }
}

// hW[r][f] = sum_k A[r][k] * W[k][f].
// Block = 256 threads = 8 waves; one 16-row M-tile per block, wave w owns
// N-tile w (F=128 -> 8 tiles). K processed in 64-wide chunks staged in LDS
// via async b128 copies, then consumed with v_wmma_f32_16x16x4_f32.
__global__ void k_gemm_wmma(const float* __restrict__ A, const float* __restrict__ W,
                            float* __restrict__ D, int K) {
  __shared__ float Al[16 * kAStride];
  __shared__ float Bl[64 * kBStride];
  int t = threadIdx.x;
  int lane = t & 31;
  int wave = t >> 5;              // nt
  int half = lane >> 4;           // lanes 16-31 hold K+2/K+3 (A rows / B k-rows)
  int mrow = lane & 15;
  int col = wave * 16 + (lane & 15);
  int mt = blockIdx.x;
  const float* arow0 = A + (size_t)mt * 16 * K;
  v8f acc = {};
  for (int k0 = 0; k0 < K; k0 += 64) {
    // Stage A chunk 16x64: thread t copies 4 floats (row t>>4, k (t&15)*4)
    {
      int row = t >> 4;
      int kk = (t & 15) * 4;
      cp16(arow0 + (size_t)row * K + k0 + kk, Al + row * kAStride + kk);
    }
    // Stage B chunk 64x128: 2048 float4s over 256 threads
#pragma unroll
    for (int j = 0; j < 8; ++j) {
      int f4 = j * 256 + t;
      int row = f4 >> 5;          // k within chunk
      int c4 = (f4 & 31) * 4;     // col
      cp16(W + (size_t)(k0 + row) * kF + c4, Bl + row * kBStride + c4);
    }
    cp_wait();
    __syncthreads();
#pragma unroll
    for (int kk = 0; kk < 64; kk += 4) {
      v2f a = *(const v2f*)(Al + mrow * kAStride + kk + 2 * half);
      v2f b;
      b.x = Bl[(kk + 2 * half) * kBStride + col];
      b.y = Bl[(kk + 2 * half + 1) * kBStride + col];
      acc = __builtin_amdgcn_wmma_f32_16x16x4_f32(false, a, false, b, (short)0, acc,
                                                  false, false);
    }
    __syncthreads();
  }
  // C/D layout: VGPR v -> row (v + 8*half), col (lane&15)
  float* drow = D + ((size_t)(mt * 16 + half * 8)) * kF + col;
#pragma unroll
  for (int v = 0; v < 8; ++v) drow[(size_t)v * kF] = acc[v];
}

__global__ void k_bias_init(float* __restrict__ a, const float* __restrict__ bias) {
  int tid = blockIdx.x * blockDim.x + threadIdx.x;   // total kM*kF
  a[tid] = bias[tid & (kF - 1)];
}

// a[g,dst,f] += norm_e * hW[g,src,f]; 128 consecutive threads share one edge.
__global__ void k_scatter(const long long* __restrict__ ei, const float* __restrict__ nrm,
                          const float* __restrict__ hW, float* __restrict__ a) {
  long long tid = (long long)blockIdx.x * blockDim.x + threadIdx.x;
  int f = (int)(tid & (kF - 1));
  long long idx = tid >> 7;
  int e = (int)(idx % kET);
  int g = (int)(idx / kET);
  if (g >= kG) return;
  int s, d;
  if (e < kE) { s = (int)ei[e]; d = (int)ei[kE + e]; }
  else        { s = d = e - kE; }
  float v = hW[((size_t)g * kL + s) * kF + f] * nrm[e];
  atomicAdd(&a[((size_t)g * kL + d) * kF + f], v);
}

__global__ void k_amax(const float* __restrict__ a, float* __restrict__ mx) {
  int tid = blockIdx.x * blockDim.x + threadIdx.x;   // total kB*kL*kF
  int f = tid & (kF - 1);
  int l = (tid >> 7) & (kL - 1);
  int b = tid >> 18;
  if (b >= kB) return;
  float m = -3.402823466e38f;
#pragma unroll
  for (int n = 0; n < kN; ++n)
    m = fmaxf(m, a[(((size_t)(b * kN + n)) * kL + l) * kF + f]);
  mx[tid] = m;
}

__global__ void k_stats_zero(float* s) {
  int i = blockIdx.x * blockDim.x + threadIdx.x;
  if (i < 4 * kF) s[i] = 0.f;
}

// blockDim = 128; thread owns channel c, accumulates a row-chunk, 2 atomics/block.
__global__ void k_stats(const float* __restrict__ t, float* __restrict__ sum,
                        float* __restrict__ ssq, int rowsPerBlock) {
  int c = threadIdx.x;
  size_t r0 = (size_t)blockIdx.x * rowsPerBlock;
  float s = 0.f, q = 0.f;
  for (int i = 0; i < rowsPerBlock; ++i) {
    float v = t[(r0 + i) * kF + c];
    s += v;
    q += v * v;
  }
  atomicAdd(&sum[c], s);
  atomicAdd(&ssq[c], q);
}

__global__ void k_finalize(const float* __restrict__ stats, const float* __restrict__ gamma,
                           const float* __restrict__ beta, float* __restrict__ scale,
                           float* __restrict__ shift) {
  int c = threadIdx.x;  // 256 threads
  float s, q, cnt;
  if (c < kF) { s = stats[c];               q = stats[kF + c];            cnt = (float)kM; }
  else        { s = stats[2 * kF + c - kF]; q = stats[3 * kF + c - kF];   cnt = (float)(kB * kL); }
  float mean = s / cnt;
  float var  = q / cnt - mean * mean;    // biased var, matches jnp.var
  float sc = rsqrtf(var + kEps) * gamma[c];
  scale[c] = sc;
  shift[c] = beta[c] - mean * sc;
}

// normalize + ReLU; channels [0,F) from a, [F,2F) from broadcast group-max.
template <bool TR>
__global__ void k_apply(const float* __restrict__ a, const float* __restrict__ mx,
                        const float* __restrict__ scale, const float* __restrict__ shift,
                        float* __restrict__ out) {
  int tid = blockIdx.x * blockDim.x + threadIdx.x;   // total kM*kC2
  int c = tid & (kC2 - 1);
  int r = tid >> 8;                                  // r = g*kL + l
  if (r >= kM) return;
  int l = r & (kL - 1);
  int g = r >> 11;
  float v;
  if (c < kF) v = a[(size_t)r * kF + c];
  else        v = mx[(((size_t)(g >> 4)) * kL + l) * kF + (c - kF)];
  v = fmaxf(v * scale[c] + shift[c], 0.f);
  if (TR) out[((size_t)g * kC2 + c) * kL + l] = v;   // final [G, 2F, L]
  else    out[(size_t)r * kC2 + c] = v;              // next block input [G, L, 2F]
}

}  // namespace

extern "C" void kernel_launch(void* const* d_in, const int* in_sizes, int n_in,
                              void* d_out, int out_size, void* d_ws, size_t ws_size,
                              hipStream_t stream) {
  const float* x = (const float*)d_in[0];
  const long long* ei = (const long long*)d_in[1];
  const float* W[3]    = {(const float*)d_in[2], (const float*)d_in[4],  (const float*)d_in[6]};
  const float* bias[3] = {(const float*)d_in[3], (const float*)d_in[5],  (const float*)d_in[7]};
  const float* gam[3]  = {(const float*)d_in[8], (const float*)d_in[10], (const float*)d_in[12]};
  const float* bet[3]  = {(const float*)d_in[9], (const float*)d_in[11], (const float*)d_in[13]};

  float* ws    = (float*)d_ws;
  float* deg   = ws;                 // kL (becomes dinv in place)
  float* nrm   = deg + kL;           // kET
  float* stats = nrm + kET;          // 4*kF: sumA, ssqA, sumM, ssqM
  float* scale = stats + 4 * kF;     // kC2
  float* shift = scale + kC2;        // kC2
  float* hbuf  = shift + kC2;        // kM * kC2
  float* hW    = hbuf + (size_t)kM * kC2;   // kM * kF
  float* abuf  = hW + (size_t)kM * kF;      // kM * kF
  float* amx   = abuf + (size_t)kM * kF;    // kB * kL * kF

  size_t needBytes =
      (size_t)(kL + kET + 4 * kF + 2 * kC2) * 4 +
      ((size_t)kM * kC2 + 2 * (size_t)kM * kF + (size_t)kB * kL * kF) * 4;
  if (ws_size < needBytes) return;

  const int T = 256;
  k_transpose_x<<<(kG * kL * 64) / T, T, 0, stream>>>(x, hbuf);
  k_deg_init<<<(kL + T - 1) / T, T, 0, stream>>>(deg);
  k_deg_acc<<<(kE + T - 1) / T, T, 0, stream>>>(ei, deg);
  k_dinv<<<(kL + T - 1) / T, T, 0, stream>>>(deg);
  k_norm<<<(kET + T - 1) / T, T, 0, stream>>>(ei, deg, nrm);

  const int Ks[3] = {64, kC2, kC2};
  for (int blk = 0; blk < 3; ++blk) {
    k_gemm_wmma<<<kM / 16, T, 0, stream>>>(hbuf, W[blk], hW, Ks[blk]);
    k_bias_init<<<(kM * kF) / T, T, 0, stream>>>(abuf, bias[blk]);
    long long scatterTotal = (long long)kG * kET * kF;
    k_scatter<<<(unsigned)(scatterTotal / T), T, 0, stream>>>(ei, nrm, hW, abuf);
    k_amax<<<(kB * kL * kF) / T, T, 0, stream>>>(abuf, amx);
    k_stats_zero<<<2, T, 0, stream>>>(stats);
    k_stats<<<512, kF, 0, stream>>>(abuf, stats, stats + kF, kM / 512);
    k_stats<<<128, kF, 0, stream>>>(amx, stats + 2 * kF, stats + 3 * kF, (kB * kL) / 128);
    k_finalize<<<1, kC2, 0, stream>>>(stats, gam[blk], bet[blk], scale, shift);
    if (blk < 2)
      k_apply<false><<<(kM * kC2) / T, T, 0, stream>>>(abuf, amx, scale, shift, hbuf);
    else
      k_apply<true><<<(kM * kC2) / T, T, 0, stream>>>(abuf, amx, scale, shift, (float*)d_out);
  }
}